// GPLoss_4037269258496
// MI455X (gfx1250) — compile-verified
//
#include <hip/hip_runtime.h>
#include <math.h>

#define EPSN 1e-12f

typedef float v2f __attribute__((ext_vector_type(2)));
typedef float v8f __attribute__((ext_vector_type(8)));

__device__ __forceinline__ float wred32(float v) {
#pragma unroll
  for (int m = 16; m > 0; m >>= 1) v += __shfl_xor(v, m, 32);
  return v;
}

// Scratch layout (floats):
//   ws[0]                 : row-cosine accumulator (pre-scaled, positive)
//   ws[1 + bc*3069 + ...] : per-(b,c) column stats
//       [0..511)dotV [511..1022)nxV [1022..1533)nrV
//       [1533..2045)dotH [2045..2557)nxH [2557..3069)nrH
#define NWS (1 + 96 * 3069)

__global__ void gp_zero(float* ws, int n) {
  int i = blockIdx.x * blockDim.x + threadIdx.x;
  if (i < n) ws[i] = 0.0f;
}

// grid = 96 images * 16 stripes; block = 128 threads = 4 waves.
// Wave owns rows [stripe*32 + w*8, +8); lane owns cols [16*lane, 16*lane+16).
__global__ __launch_bounds__(128) void gp_main(const float* __restrict__ X,
                                               const float* __restrict__ R,
                                               float* __restrict__ ws) {
  __shared__ __align__(16) float stage[4][3][2][512];  // 48 KB
  __shared__ float colbuf[3069];                        // 12 KB

  const int tid    = threadIdx.x;
  const int lane   = tid & 31;
  const int w      = tid >> 5;
  const int bc     = blockIdx.x >> 4;
  const int stripe = blockIdx.x & 15;
  const int rowBeg = stripe * 32 + w * 8;
  const int rowEnd = rowBeg + 8;

  for (int i = tid; i < 3069; i += 128) colbuf[i] = 0.0f;
  __syncthreads();

  const size_t img = (size_t)bc * (512u * 512u);
  const float* gX = X + img;
  const float* gR = R + img;

  float cdV[16], cxV[16], crV[16], cdH[16], cxH[16], crH[16];
#pragma unroll
  for (int j = 0; j < 16; ++j) {
    cdV[j] = 0.f; cxV[j] = 0.f; crV[j] = 0.f;
    cdH[j] = 0.f; cxH[j] = 0.f; crH[j] = 0.f;
  }

  float accRow = 0.0f;

  // 8 async DMA ops per row pair; IOFFSET advances LDS and global together.
  auto issueRow = [&](int row, int s) {
    const float* px = gX + (size_t)row * 512 + lane * 4;
    const float* pr = gR + (size_t)row * 512 + lane * 4;
    unsigned lx = (unsigned)(size_t)&stage[w][s][0][lane * 4];
    unsigned lr = (unsigned)(size_t)&stage[w][s][1][lane * 4];
    asm volatile(
        "global_load_async_to_lds_b128 %0, %2, off\n\t"
        "global_load_async_to_lds_b128 %0, %2, off offset:512\n\t"
        "global_load_async_to_lds_b128 %0, %2, off offset:1024\n\t"
        "global_load_async_to_lds_b128 %0, %2, off offset:1536\n\t"
        "global_load_async_to_lds_b128 %1, %3, off\n\t"
        "global_load_async_to_lds_b128 %1, %3, off offset:512\n\t"
        "global_load_async_to_lds_b128 %1, %3, off offset:1024\n\t"
        "global_load_async_to_lds_b128 %1, %3, off offset:1536"
        :
        : "v"(lx), "v"(lr), "v"(px), "v"(pr)
        : "memory");
  };

  issueRow(rowBeg, 0);
  issueRow(rowBeg + 1, 1);

  for (int i = rowBeg; i < rowEnd; ++i) {
    const int idx = i - rowBeg;
    const int sc = idx % 3;
    const int sn = (idx + 1) % 3;
    const int sp = (idx + 2) % 3;
    const int pf = i + 2;
    const bool doPf    = (pf <= rowEnd) && (pf <= 511);
    const bool hasNext = (i + 1) < 512;

    if (doPf) {
      asm volatile("s_wait_dscnt 0" ::: "memory");
      issueRow(pf, sp);
      asm volatile("s_wait_asynccnt 8" ::: "memory");
    } else {
      asm volatile("s_wait_asynccnt 0" ::: "memory");
    }

    const float* sxc = &stage[w][sc][0][0];
    const float* src = &stage[w][sc][1][0];

    float xc[16], rc[16];
#pragma unroll
    for (int q = 0; q < 4; ++q) {
      float4 a = ((const float4*)(sxc + lane * 16))[q];
      xc[4 * q + 0] = a.x; xc[4 * q + 1] = a.y;
      xc[4 * q + 2] = a.z; xc[4 * q + 3] = a.w;
      float4 b = ((const float4*)(src + lane * 16))[q];
      rc[4 * q + 0] = b.x; rc[4 * q + 1] = b.y;
      rc[4 * q + 2] = b.z; rc[4 * q + 3] = b.w;
    }

    const int bidx = (lane < 31) ? (lane * 16 + 16) : 511;
    float xRb = sxc[bidx];
    float rRb = src[bidx];

    float rdV = 0.f, rxV = 0.f, rrV = 0.f;
    float rdH = 0.f, rxH = 0.f, rrH = 0.f;

    const float* snx = &stage[w][sn][0][0];
    const float* snr = &stage[w][sn][1][0];
#pragma unroll
    for (int q = 0; q < 4; ++q) {
      float xnA[4] = {0.f, 0.f, 0.f, 0.f};
      float rnA[4] = {0.f, 0.f, 0.f, 0.f};
      if (hasNext) {
        float4 xn4 = ((const float4*)(snx + lane * 16))[q];
        float4 rn4 = ((const float4*)(snr + lane * 16))[q];
        xnA[0] = xn4.x; xnA[1] = xn4.y; xnA[2] = xn4.z; xnA[3] = xn4.w;
        rnA[0] = rn4.x; rnA[1] = rn4.y; rnA[2] = rn4.z; rnA[3] = rn4.w;
      }
#pragma unroll
      for (int t = 0; t < 4; ++t) {
        const int j = q * 4 + t;
        const float xcv = xc[j], rcv = rc[j];
        const float xnx = (j < 15) ? xc[j + 1] : xRb;
        const float rnx = (j < 15) ? rc[j + 1] : rRb;
        if ((j < 15) || (lane < 31)) {
          float gx = xcv - xnx, gr = rcv - rnx;
          float pd = gx * gr, pxx = gx * gx, prr = gr * gr;
          rdV += pd; rxV += pxx; rrV += prr;
          cdV[j] += pd; cxV[j] += pxx; crV[j] += prr;
        }
        if (hasNext) {
          float gx = xcv - xnA[t], gr = rcv - rnA[t];
          float pd = gx * gr, pxx = gx * gx, prr = gr * gr;
          rdH += pd; rxH += pxx; rrH += prr;
          cdH[j] += pd; cxH[j] += pxx; crH[j] += prr;
        }
      }
    }

    rdV = wred32(rdV); rxV = wred32(rxV); rrV = wred32(rrV);
    float cosV = rdV / (fmaxf(sqrtf(rxV), EPSN) * fmaxf(sqrtf(rrV), EPSN));
    accRow += cosV * (1.0f / (512.0f * 32.0f));
    if (hasNext) {
      rdH = wred32(rdH); rxH = wred32(rxH); rrH = wred32(rrH);
      float cosH = rdH / (fmaxf(sqrtf(rxH), EPSN) * fmaxf(sqrtf(rrH), EPSN));
      accRow += cosH * (1.0f / (511.0f * 32.0f));
    }
  }

  if (lane == 0) atomicAdd(&ws[0], accRow);

  const int c0 = lane * 16;
#pragma unroll
  for (int j = 0; j < 16; ++j) {
    const int c = c0 + j;
    if (c < 511) {
      atomicAdd(&colbuf[c], cdV[j]);
      atomicAdd(&colbuf[511 + c], cxV[j]);
      atomicAdd(&colbuf[1022 + c], crV[j]);
    }
    atomicAdd(&colbuf[1533 + c], cdH[j]);
    atomicAdd(&colbuf[2045 + c], cxH[j]);
    atomicAdd(&colbuf[2557 + c], crH[j]);
  }
  __syncthreads();
  float* g = ws + 1 + (size_t)bc * 3069;
  for (int i = tid; i < 3069; i += 128) atomicAdd(&g[i], colbuf[i]);
}

__global__ __launch_bounds__(256) void gp_final(const float* __restrict__ ws,
                                                float* __restrict__ out) {
  __shared__ float red[256];
  const int tid = threadIdx.x;
  float acc = 0.0f;

  for (int idx = tid; idx < 96 * 511; idx += 256) {
    const int b = idx / 511;
    const int c = idx - b * 511;
    const float* g = ws + 1 + (size_t)b * 3069;
    float d = g[c], nx = g[511 + c], nr = g[1022 + c];
    acc += (d / (fmaxf(sqrtf(nx), EPSN) * fmaxf(sqrtf(nr), EPSN))) *
           (1.0f / (511.0f * 32.0f));
  }
  for (int idx = tid; idx < 96 * 512; idx += 256) {
    const int b = idx >> 9;
    const int c = idx & 511;
    const float* g = ws + 1 + (size_t)b * 3069 + 1533;
    float d = g[c], nx = g[512 + c], nr = g[1024 + c];
    acc += (d / (fmaxf(sqrtf(nx), EPSN) * fmaxf(sqrtf(nr), EPSN))) *
           (1.0f / (512.0f * 32.0f));
  }

  red[tid] = acc;
  __syncthreads();
  if (tid < 32) {
    float p = 0.0f;
#pragma unroll
    for (int k = 0; k < 8; ++k) p += red[tid + 32 * k];
    // 32-lane sum via V_WMMA_F32_16X16X4_F32 with all-ones B:
    // A[m,k] = {p(m), 0, p(m+16), 0}  =>  D[m,*] = p(m) + p(m+16)
    v2f A;  A[0] = p;     A[1] = 0.0f;
    v2f Bv; Bv[0] = 1.0f; Bv[1] = 1.0f;
    v8f C = {};
    v8f D = __builtin_amdgcn_wmma_f32_16x16x4_f32(false, A, false, Bv,
                                                  (short)0, C, false, false);
    float s = D[0] + D[1] + D[2] + D[3] + D[4] + D[5] + D[6] + D[7];
    s += __shfl_xor(s, 16, 32);
    if (tid == 0) out[0] = -(s + ws[0]);
  }
}

extern "C" void kernel_launch(void* const* d_in, const int* in_sizes, int n_in,
                              void* d_out, int out_size, void* d_ws,
                              size_t ws_size, hipStream_t stream) {
  (void)in_sizes; (void)n_in; (void)out_size; (void)ws_size;
  const float* X = (const float*)d_in[0];
  const float* R = (const float*)d_in[1];
  float* ws = (float*)d_ws;
  float* out = (float*)d_out;

  gp_zero<<<(NWS + 255) / 256, 256, 0, stream>>>(ws, NWS);
  gp_main<<<96 * 16, 128, 0, stream>>>(X, R, ws);
  gp_final<<<1, 256, 0, stream>>>(ws, out);
}